// EnetGnn_64931315581206
// MI455X (gfx1250) — compile-verified
//
#include <hip/hip_runtime.h>
#include <math.h>

// Problem constants (from reference setup_inputs)
#define N_B   4
#define C_F   3
#define HW    9216            // 96*96
#define KNN   16
#define QTILE 16              // queries per workgroup in knn
#define CHUNK 2304            // candidates processed per LDS chunk (9216/4)
#define NCHUNK (HW / CHUNK)   // 4
#define TILES_PER_CHUNK (CHUNK / 16)  // 144
#define SROW  2324            // padded LDS row stride in floats (mult of 4, bank-tuned)
#define SMEM_BYTES (QTILE * SROW * 4) // 148736 bytes dynamic LDS (<320KB WGP)

typedef __attribute__((ext_vector_type(2))) float v2f;
typedef __attribute__((ext_vector_type(4))) float v4f;
typedef __attribute__((ext_vector_type(8))) float v8f;

// ---------------------------------------------------------------------------
// Prep: proj_3d [N,HW,3] -> padded pts [N,HW,4] + squared norms [N,HW]
__global__ __launch_bounds__(256)
void prep_pts_kernel(const float* __restrict__ proj,
                     float* __restrict__ pts,
                     float* __restrict__ norms, int total) {
    int t = blockIdx.x * blockDim.x + threadIdx.x;
    if (t >= total) return;
    const float* p = proj + (size_t)t * 3;
    float x = p[0], y = p[1], z = p[2];
    float* o = pts + (size_t)t * 4;
    o[0] = x; o[1] = y; o[2] = z; o[3] = 0.0f;
    norms[t] = x * x + y * y + z * z;
}

// Prep: cnn [N,C,HW] -> padded h [N,HW,4]
__global__ __launch_bounds__(256)
void prep_h_kernel(const float* __restrict__ cnn,
                   float* __restrict__ h, int total) {
    int t = blockIdx.x * blockDim.x + threadIdx.x;
    if (t >= total) return;
    int n = t / HW, p = t % HW;
    const float* base = cnn + (size_t)n * 3 * HW + p;
    float* o = h + (size_t)t * 4;
    o[0] = base[0];
    o[1] = base[HW];
    o[2] = base[2 * HW];
    o[3] = 0.0f;
}

// Final: padded h [N,HW,4] -> out [N,C,HW]
__global__ __launch_bounds__(256)
void writeout_kernel(const float* __restrict__ h,
                     float* __restrict__ out, int total) {
    int t = blockIdx.x * blockDim.x + threadIdx.x;
    if (t >= total) return;
    int n = t / HW, p = t % HW;
    const float* s = h + (size_t)t * 4;
    float* o = out + (size_t)n * 3 * HW + p;
    o[0]      = s[0];
    o[HW]     = s[1];
    o[2 * HW] = s[2];
}

// ---------------------------------------------------------------------------
// KNN: per workgroup, 16 queries vs all HW candidates of one batch.
// score(m,n) = |x_n|^2 - 2 x_m . x_n   (row-constant |x_m|^2 dropped:
// top-k ordering invariant). One V_WMMA_F32_16X16X4_F32 per 16x16 tile:
//   A = -2 * queries (16x4, K padded), B = candidates (4x16), C = |x_n|^2.
__global__ __launch_bounds__(256)
void knn_kernel(const float* __restrict__ pts,    // [N,HW,4]
                const float* __restrict__ norms,  // [N,HW]
                int* __restrict__ knn_out)        // [N,HW,KNN]
{
    extern __shared__ char smem_raw[];
    float* S = (float*)smem_raw;

    const int tid   = threadIdx.x;
    const int lane  = tid & 31;
    const int wave  = tid >> 5;        // 0..7
    const int qhalf = lane >> 4;       // 0 or 1: coord pair select
    const int qlane = lane & 15;       // point within 16-tile

    const int block = blockIdx.x;
    const int n     = block / (HW / QTILE);
    const int qbase = (block % (HW / QTILE)) * QTILE;

    const float* ptsN  = pts   + (size_t)n * HW * 4;
    const float* normN = norms + (size_t)n * HW;

    // A fragment (32-bit A 16x4 layout): lane l holds query l%16,
    // coords K=(l/16)*2 and K+1. Fold -2 scale into A, load once.
    v2f afrag;
    {
        const float* pq = ptsN + (size_t)(qbase + qlane) * 4 + qhalf * 2;
        afrag.x = -2.0f * pq[0];
        afrag.y = -2.0f * pq[1];
    }

    // Register-resident sorted top-16 (ascending). Selector thread
    // (q = tid>>4, s = tid&15) scans column quads [4s + 64i, +4).
    float bs[KNN];
    int   bi[KNN];
#pragma unroll
    for (int j = 0; j < KNN; ++j) { bs[j] = 3.0e38f; bi[j] = -1; }

    const int selq = tid >> 4;
    const int sels = tid & 15;

    for (int ch = 0; ch < NCHUNK; ++ch) {
        const int cbase = ch * CHUNK;
        __syncthreads();   // previous chunk's selection done before overwrite

        // ---- WMMA fill phase: wave-uniform trip count, EXEC all ones ----
        for (int t = wave; t < TILES_PER_CHUNK; t += 8) {
            const int c0 = cbase + t * 16;
            // Branchless WGP-scope prefetch of the next tile's candidate
            // points; target clamped in-bounds (WGP prefetch is treated as
            // non-speculative). Lowers to global_prefetch_b8.
            {
                const int pt = (t + 8 < TILES_PER_CHUNK) ? (t + 8) : t;
                __builtin_prefetch(ptsN + (size_t)(cbase + pt * 16 + qlane) * 4, 0, 3);
            }
            const float* pc = ptsN + (size_t)(c0 + qlane) * 4 + qhalf * 2;
            v2f bfrag;
            bfrag.x = pc[0];
            bfrag.y = pc[1];
            const float cn = normN[c0 + qlane];   // |x_n|^2, constant per column
            v8f acc;
#pragma unroll
            for (int v = 0; v < 8; ++v) acc[v] = cn;
            v8f d = __builtin_amdgcn_wmma_f32_16x16x4_f32(
                false, afrag, false, bfrag, (short)0, acc, false, false);
            // D layout: lane l, VGPR v -> row M = v + 8*(l/16), col N = l%16
            const int col   = t * 16 + qlane;
            const int rbase = qhalf * 8;
#pragma unroll
            for (int v = 0; v < 8; ++v)
                S[(rbase + v) * SROW + col] = d[v];
        }
        __syncthreads();

        // ---- selection: b128 quad loads + quick reject, then unrolled
        //      insertion into the sorted register list ----
        const float* Sq = S + selq * SROW;
        for (int c0 = sels * 4; c0 < CHUNK; c0 += 64) {
            v4f v = *(const v4f*)(Sq + c0);      // ds_load_b128
            float mn = fminf(fminf(v.x, v.y), fminf(v.z, v.w));
            if (mn < bs[KNN - 1]) {
#pragma unroll
                for (int e = 0; e < 4; ++e) {
                    float val = v[e];
                    if (val < bs[KNN - 1]) {
                        float cv = val; int ci = cbase + c0 + e;
#pragma unroll
                        for (int j = 0; j < KNN; ++j) {
                            if (cv < bs[j]) {
                                float ts = bs[j]; int ti = bi[j];
                                bs[j] = cv; bi[j] = ci;
                                cv = ts; ci = ti;
                            }
                        }
                    }
                }
            }
        }
    }

    // ---- merge: 16 sorted register lists per query -> global top-16.
    // The 16 selector threads of a query are a contiguous 16-lane group,
    // so a width-16 shfl_xor butterfly finds the global min head each
    // round; the owner pops (unrolled shift keeps everything in VGPRs).
    int* outp = knn_out + ((size_t)n * HW + qbase + selq) * KNN;
    for (int r = 0; r < KNN; ++r) {
        float v  = bs[0];
        int   id = bi[0];
        int   ow = sels;
#pragma unroll
        for (int off = 8; off >= 1; off >>= 1) {
            float ov  = __shfl_xor(v, off, 16);
            int   oid = __shfl_xor(id, off, 16);
            int   oo  = __shfl_xor(ow, off, 16);
            // total order (score, owner) so all lanes converge identically
            if (ov < v || (ov == v && oo < ow)) { v = ov; id = oid; ow = oo; }
        }
        if (ow == sels) {   // winner pops its head
#pragma unroll
            for (int j = 0; j < KNN - 1; ++j) { bs[j] = bs[j + 1]; bi[j] = bi[j + 1]; }
            bs[KNN - 1] = 3.0e38f; bi[KNN - 1] = -1;
        }
        if (sels == r) outp[r] = id;
    }
}

// ---------------------------------------------------------------------------
// One GNN iteration: gather 16 neighbors, 2-layer 3->3 MLP + PReLU, mean,
// concat(h, m) @ qW^T + qb, ReLU. Writes padded h_new + its squared norms.
__global__ __launch_bounds__(256)
void gnn_kernel(const float* __restrict__ hin,   // [N,HW,4]
                const int*   __restrict__ knn,   // [N,HW,KNN] (batch-local idx)
                const float* __restrict__ gW,    // [2,3,3]
                const float* __restrict__ gB,    // [2,3]
                const float* __restrict__ gA,    // [2]
                const float* __restrict__ qW,    // [3,6]
                const float* __restrict__ qB,    // [3]
                float* __restrict__ hout,        // [N,HW,4]
                float* __restrict__ normout)     // [N,HW]
{
    int t = blockIdx.x * blockDim.x + threadIdx.x;
    if (t >= N_B * HW) return;
    int n = t / HW;
    const float* hN = hin + (size_t)n * HW * 4;

    float W0[9], W1[9], B0[3], B1[3];
#pragma unroll
    for (int i = 0; i < 9; ++i) { W0[i] = gW[i]; W1[i] = gW[9 + i]; }
#pragma unroll
    for (int i = 0; i < 3; ++i) { B0[i] = gB[i]; B1[i] = gB[3 + i]; }
    const float A0 = gA[0], A1 = gA[1];

    const float* hs = hin + (size_t)t * 4;
    const float sx = hs[0], sy = hs[1], sz = hs[2];

    const int* nb = knn + (size_t)t * KNN;
    float m0 = 0.f, m1 = 0.f, m2 = 0.f;
#pragma unroll 4
    for (int kk = 0; kk < KNN; ++kk) {
        const float* pj = hN + (size_t)nb[kk] * 4;
        float x = pj[0], y = pj[1], z = pj[2];
        // layer 0 + PReLU
        float u0 = W0[0]*x + W0[1]*y + W0[2]*z + B0[0];
        float u1 = W0[3]*x + W0[4]*y + W0[5]*z + B0[1];
        float u2 = W0[6]*x + W0[7]*y + W0[8]*z + B0[2];
        u0 = (u0 >= 0.f) ? u0 : A0 * u0;
        u1 = (u1 >= 0.f) ? u1 : A0 * u1;
        u2 = (u2 >= 0.f) ? u2 : A0 * u2;
        // layer 1 + PReLU
        float w0 = W1[0]*u0 + W1[1]*u1 + W1[2]*u2 + B1[0];
        float w1 = W1[3]*u0 + W1[4]*u1 + W1[5]*u2 + B1[1];
        float w2 = W1[6]*u0 + W1[7]*u1 + W1[8]*u2 + B1[2];
        w0 = (w0 >= 0.f) ? w0 : A1 * w0;
        w1 = (w1 >= 0.f) ? w1 : A1 * w1;
        w2 = (w2 >= 0.f) ? w2 : A1 * w2;
        m0 += w0; m1 += w1; m2 += w2;
    }
    m0 *= (1.0f / KNN); m1 *= (1.0f / KNN); m2 *= (1.0f / KNN);

    float o[3];
#pragma unroll
    for (int i = 0; i < 3; ++i) {
        const float* qw = qW + i * 6;
        float v = qw[0]*sx + qw[1]*sy + qw[2]*sz
                + qw[3]*m0 + qw[4]*m1 + qw[5]*m2 + qB[i];
        o[i] = (v > 0.f) ? v : 0.f;
    }
    float* ho = hout + (size_t)t * 4;
    ho[0] = o[0]; ho[1] = o[1]; ho[2] = o[2]; ho[3] = 0.0f;
    normout[t] = o[0]*o[0] + o[1]*o[1] + o[2]*o[2];
}

// ---------------------------------------------------------------------------
extern "C" void kernel_launch(void* const* d_in, const int* in_sizes, int n_in,
                              void* d_out, int out_size, void* d_ws, size_t ws_size,
                              hipStream_t stream) {
    (void)in_sizes; (void)n_in; (void)out_size; (void)ws_size;

    const float* cnn  = (const float*)d_in[0];
    const float* proj = (const float*)d_in[1];
    const float* gW   = (const float*)d_in[2];
    const float* gB   = (const float*)d_in[3];
    const float* gA   = (const float*)d_in[4];
    const float* qW   = (const float*)d_in[5];
    const float* qB   = (const float*)d_in[6];
    // gnn_iterations(=3) and k(=16) are fixed by the reference setup
    float* out = (float*)d_out;

    // workspace carve-up (~4.1 MB)
    float* pts   = (float*)d_ws;                 // N*HW*4 floats
    float* hA    = pts   + (size_t)N_B * HW * 4;
    float* hB    = hA    + (size_t)N_B * HW * 4;
    float* norms = hB    + (size_t)N_B * HW * 4; // N*HW floats
    int*   idx   = (int*)(norms + (size_t)N_B * HW); // N*HW*KNN ints

    const int total = N_B * HW;
    dim3 blk(256);
    dim3 gpt((total + 255) / 256);
    dim3 gknn(N_B * (HW / QTILE));  // 2304 workgroups

    prep_pts_kernel<<<gpt, blk, 0, stream>>>(proj, pts, norms, total);
    prep_h_kernel<<<gpt, blk, 0, stream>>>(cnn, hA, total);

    // iter 0: knn(proj_3d), gnn
    knn_kernel<<<gknn, blk, SMEM_BYTES, stream>>>(pts, norms, idx);
    gnn_kernel<<<gpt, blk, 0, stream>>>(hA, idx, gW, gB, gA, qW, qB, hB, norms);
    // iter 1: knn(h), gnn
    knn_kernel<<<gknn, blk, SMEM_BYTES, stream>>>(hB, norms, idx);
    gnn_kernel<<<gpt, blk, 0, stream>>>(hB, idx, gW, gB, gA, qW, qB, hA, norms);
    // iter 2: knn(h), gnn
    knn_kernel<<<gknn, blk, SMEM_BYTES, stream>>>(hA, norms, idx);
    gnn_kernel<<<gpt, blk, 0, stream>>>(hA, idx, gW, gB, gA, qW, qB, hB, norms);

    writeout_kernel<<<gpt, blk, 0, stream>>>(hB, out, total);
}